// Loss_Synonymy_34213709479937
// MI455X (gfx1250) — compile-verified
//
#include <hip/hip_runtime.h>
#include <hip/hip_bf16.h>
#include <cmath>

typedef __attribute__((ext_vector_type(2))) float v2f;
typedef __attribute__((ext_vector_type(8))) float v8f;

#define DIM 4096
#define ROWS_PER_BLOCK 16
#define THREADS 256

// Each block: 16 rows. Thread t = (row t>>4, colgroup t&15).
// Streams both matrices with coalesced float4 loads, accumulates sum of
// squared diffs per (row, colgroup) in registers, reduces the 16x16 partial
// matrix per-row with V_WMMA_F32_16X16X4_F32 against an all-ones B, then
// applies sqrt/tanh/label-sign/relu and emits one partial per block.
__global__ __launch_bounds__(THREADS)
void synonymy_main(const float* __restrict__ s1,
                   const float* __restrict__ s2,
                   const int*   __restrict__ labels,
                   float*       __restrict__ partials)
{
    __shared__ float P[ROWS_PER_BLOCK][17];   // padded: stride 17 (coprime w/ banks)

    const int tid = threadIdx.x;
    const int r   = tid >> 4;      // row within tile: 0..15
    const int cg  = tid & 15;      // column group:    0..15
    const long row = (long)blockIdx.x * ROWS_PER_BLOCK + r;

    const float* __restrict__ p1 = s1 + row * (long)DIM;
    const float* __restrict__ p2 = s2 + row * (long)DIM;

    // 16 threads per row, float4 each -> 64 contiguous floats per row-iter.
    float acc = 0.0f;
    #pragma unroll 4
    for (int i = 0; i < DIM / 64; ++i) {
        const int col = (cg << 2) + (i << 6);
        const float4 a = *(const float4*)(p1 + col);
        const float4 b = *(const float4*)(p2 + col);
        const float dx = a.x - b.x;
        const float dy = a.y - b.y;
        const float dz = a.z - b.z;
        const float dw = a.w - b.w;
        acc = fmaf(dx, dx, acc);
        acc = fmaf(dy, dy, acc);
        acc = fmaf(dz, dz, acc);
        acc = fmaf(dw, dw, acc);
    }

    P[r][cg] = acc;
    __syncthreads();

    // Wave 0 only (EXEC all-ones within the wave, as WMMA requires).
    if (tid < 32) {
        const int lane = tid;
        const int m    = lane & 15;          // A-matrix row M
        const int koff = (lane >> 4) << 1;   // lanes 16-31 carry K=2,3

        v8f c = {};
        const v2f bones = {1.0f, 1.0f};      // B = 4x16 ones -> row sums
        #pragma unroll
        for (int q = 0; q < 4; ++q) {        // K tiles 0..3 -> K=16 total
            v2f a;
            a.x = P[m][(q << 2) + koff + 0];
            a.y = P[m][(q << 2) + koff + 1];
            c = __builtin_amdgcn_wmma_f32_16x16x4_f32(
                    /*neg_a=*/false, a, /*neg_b=*/false, bones,
                    /*c_mod=*/(short)0, c, /*reuse_a=*/false, /*reuse_b=*/false);
        }

        // C layout: VGPR j, lanes 0-15 -> row M=j (any N); lanes 16-31 -> M=j+8.
        float local = 0.0f;
        if ((lane & 15) == 0) {
            const int mbase = (lane >> 4) << 3;  // 0 or 8
            const int lbase = blockIdx.x * ROWS_PER_BLOCK + mbase;
            #pragma unroll
            for (int j = 0; j < 8; ++j) {
                const float ss   = c[j];
                const float dist = sqrtf(ss);
                const float th   = tanhf(dist);
                const int   lab  = labels[lbase + j];
                const float sg   = (lab == 1) ? -th : th;
                local += fmaxf(0.0f, 1.0f + sg);
            }
        }
        local += __shfl_xor(local, 16);          // combine lane 0 + lane 16
        if (lane == 0) partials[blockIdx.x] = local;
    }
}

// Deterministic fixed-tree reduction of the per-block partials.
__global__ __launch_bounds__(THREADS)
void reduce_partials(const float* __restrict__ partials,
                     float* __restrict__ out, int n)
{
    __shared__ float sm[THREADS];
    const int tid = threadIdx.x;
    float s = 0.0f;
    for (int i = tid; i < n; i += THREADS) s += partials[i];
    sm[tid] = s;
    __syncthreads();
    #pragma unroll
    for (int off = THREADS / 2; off > 0; off >>= 1) {
        if (tid < off) sm[tid] += sm[tid + off];
        __syncthreads();
    }
    if (tid == 0) out[0] = sm[0];
}

extern "C" void kernel_launch(void* const* d_in, const int* in_sizes, int n_in,
                              void* d_out, int out_size, void* d_ws, size_t ws_size,
                              hipStream_t stream)
{
    const float* s1     = (const float*)d_in[0];
    const float* s2     = (const float*)d_in[1];
    const int*   labels = (const int*)d_in[2];
    float*       out    = (float*)d_out;
    float*       parts  = (float*)d_ws;

    const int B      = in_sizes[2];            // 16384 rows
    const int blocks = B / ROWS_PER_BLOCK;     // 1024 blocks

    synonymy_main<<<blocks, THREADS, 0, stream>>>(s1, s2, labels, parts);
    reduce_partials<<<1, THREADS, 0, stream>>>(parts, out, blocks);
}